// Decoder_1692217114985
// MI455X (gfx1250) — compile-verified
//
#include <hip/hip_runtime.h>
#include <hip/hip_bf16.h>

typedef __attribute__((ext_vector_type(2))) float v2f;
typedef __attribute__((ext_vector_type(8))) float v8f;

static __device__ __forceinline__ v8f wmma4(v2f a, v2f b, v8f c) {
    return __builtin_amdgcn_wmma_f32_16x16x4_f32(false, a, false, b, (short)0, c, false, false);
}

static __device__ __forceinline__ float apply_act(float x, int act) {
    if (act == 1) return tanhf(x);
    if (act == 2) return 1.0f / (1.0f + expf(-x));
    return x;
}

// ---------------------------------------------------------------------------
// Specialized fp32 WMMA GEMM for M=64 (4 M-tiles per wave), no masking:
//   C[64 x N] = act(A[64 x K] @ B[K x N] + bias)
// One wave per 16-wide column tile; 4 accumulators reuse each B fragment 4x
// (cuts B traffic 4x and gives 4 independent WMMAs per load pair).
// Requires: N % 16 == 0, K % 4 == 0.
// ---------------------------------------------------------------------------
__global__ __launch_bounds__(256) void gemm_wmma_f32_m64(
    const float* __restrict__ A, const float* __restrict__ B,
    const float* __restrict__ bias, float* __restrict__ C,
    int N, int K, int act)
{
    const int wave   = (blockIdx.x * blockDim.x + threadIdx.x) >> 5;
    const int lane   = threadIdx.x & 31;
    const int tilesN = N >> 4;
    if (wave >= tilesN) return;                 // wave-uniform exit

    const int n    = wave * 16 + (lane & 15);
    const int koff = (lane >> 4) << 1;          // 0 or 2
    const float* Ab = A + (lane & 15) * K + koff;
    const float* Bb = B + n + (size_t)koff * N;

    v8f acc[4] = {v8f{}, v8f{}, v8f{}, v8f{}};

#pragma unroll 2
    for (int k0 = 0; k0 < K; k0 += 4) {
        v2f b;
        b.x = Bb[(size_t)k0 * N];
        b.y = Bb[(size_t)(k0 + 1) * N];
        const v2f a0 = *(const v2f*)(Ab + k0);
        const v2f a1 = *(const v2f*)(Ab + 16 * K + k0);
        const v2f a2 = *(const v2f*)(Ab + 32 * K + k0);
        const v2f a3 = *(const v2f*)(Ab + 48 * K + k0);
        acc[0] = wmma4(a0, b, acc[0]);
        acc[1] = wmma4(a1, b, acc[1]);
        acc[2] = wmma4(a2, b, acc[2]);
        acc[3] = wmma4(a3, b, acc[3]);
    }

    const float bv    = (bias != nullptr) ? bias[n] : 0.0f;
    const int rowhalf = (lane >> 4) << 3;       // 0 or 8
#pragma unroll
    for (int t = 0; t < 4; ++t)
#pragma unroll
        for (int v = 0; v < 8; ++v)
            C[(size_t)(t * 16 + rowhalf + v) * N + n] = apply_act(acc[t][v] + bv, act);
}

// ---------------------------------------------------------------------------
// Generic fp32 WMMA GEMM with BRANCHLESS edge masking (only used for the two
// tiny FF layers): clamp index, load unconditionally, cndmask value to zero.
//   C[M x N] = act(A[M x K] @ B[Kreal x Nreal](ldb) + bias), N%16==0, K%4==0
// ---------------------------------------------------------------------------
__global__ __launch_bounds__(256) void gemm_wmma_f32_masked(
    const float* __restrict__ A, const float* __restrict__ B,
    const float* __restrict__ bias, float* __restrict__ C,
    int M, int N, int K, int Nreal, int Kreal, int ldb, int act)
{
    const int wave   = (blockIdx.x * blockDim.x + threadIdx.x) >> 5;
    const int lane   = threadIdx.x & 31;
    const int tilesN = N >> 4;
    const int tileM  = wave / tilesN;
    const int tileN  = wave - tileM * tilesN;
    if (tileM * 16 >= M) return;                // wave-uniform exit

    const int m    = tileM * 16 + (lane & 15);
    const int n    = tileN * 16 + (lane & 15);
    const int koff = (lane >> 4) << 1;
    const bool nok = (n < Nreal);
    const int nc   = nok ? n : 0;               // safe column

    v8f acc = {};
    for (int k0 = 0; k0 < K; k0 += 4) {
        const int ka  = k0 + koff;
        const int kc0 = (ka     < Kreal) ? ka     : 0;   // safe rows
        const int kc1 = (ka + 1 < Kreal) ? ka + 1 : 0;
        const v2f a   = *(const v2f*)(&A[m * K + ka]);
        const float bx = B[kc0 * ldb + nc];
        const float by = B[kc1 * ldb + nc];
        v2f b;
        b.x = (nok && (ka     < Kreal)) ? bx : 0.0f;     // branchless select
        b.y = (nok && (ka + 1 < Kreal)) ? by : 0.0f;
        acc = wmma4(a, b, acc);
    }

    const float bv    = (bias != nullptr && nok) ? bias[n] : 0.0f;
    const int rowbase = tileM * 16 + ((lane >> 4) << 3);
#pragma unroll
    for (int v = 0; v < 8; ++v)
        C[(rowbase + v) * N + n] = apply_act(acc[v] + bv, act);
}

// ---------------------------------------------------------------------------
// kval kernel: per (point n, center k) pair evaluate the two 2->20->20->1
// ReLU MLPs, masked to the 0.15 window. Writes Bm[k][n*2+f] (coalesced float2)
// -- exactly the B layout of the final GEMM. ~96% of pairs skip the MLP.
// ---------------------------------------------------------------------------
__global__ __launch_bounds__(256) void kval_kernel(
    const float* __restrict__ grid,
    const float* __restrict__ k_w1, const float* __restrict__ k_b1,
    const float* __restrict__ k_w2, const float* __restrict__ k_b2,
    const float* __restrict__ k_w3, const float* __restrict__ k_b3,
    float* __restrict__ Bm)   // [400][4096]
{
    __shared__ float sw1[2][2][20];
    __shared__ float sb1[2][20];
    __shared__ float sw2[2][20][20];
    __shared__ float sb2[2][20];
    __shared__ float sw3[2][20];
    __shared__ float sb3[2];

    for (int i = threadIdx.x; i < 80;  i += 256) (&sw1[0][0][0])[i] = k_w1[i];
    for (int i = threadIdx.x; i < 40;  i += 256) (&sb1[0][0])[i]    = k_b1[i];
    for (int i = threadIdx.x; i < 800; i += 256) (&sw2[0][0][0])[i] = k_w2[i];
    for (int i = threadIdx.x; i < 40;  i += 256) (&sb2[0][0])[i]    = k_b2[i];
    for (int i = threadIdx.x; i < 40;  i += 256) (&sw3[0][0])[i]    = k_w3[i];
    if (threadIdx.x < 2) sb3[threadIdx.x] = k_b3[threadIdx.x];
    __syncthreads();

    const int tid = blockIdx.x * 256 + threadIdx.x;
    const int n = tid & 2047;        // point index (fastest -> coalesced)
    const int k = tid >> 11;         // center index (uniform per wave)
    if (k >= 400) return;

    const float gx = grid[n * 2 + 0];
    const float gy = grid[n * 2 + 1];
    const float cx = (float)(k / 20) * 0.05f;
    const float cy = (float)(k % 20) * 0.05f;
    const float lx = gx - cx;
    const float ly = gy - cy;

    float2 r = make_float2(0.0f, 0.0f);
    const bool inside = (lx >= 0.0f) && (lx <= 0.15f) && (ly >= 0.0f) && (ly <= 0.15f);
    if (inside) {
        float out[2];
#pragma unroll
        for (int f = 0; f < 2; ++f) {
            float h1[20];
#pragma unroll
            for (int j = 0; j < 20; ++j) {
                float t = fmaf(lx, sw1[f][0][j], fmaf(ly, sw1[f][1][j], sb1[f][j]));
                h1[j] = fmaxf(t, 0.0f);
            }
            float kv = sb3[f];
#pragma unroll 4
            for (int g = 0; g < 20; ++g) {
                float s = sb2[f][g];
#pragma unroll
                for (int j = 0; j < 20; ++j) s = fmaf(h1[j], sw2[f][j][g], s);
                kv = fmaf(fmaxf(s, 0.0f), sw3[f][g], kv);
            }
            out[f] = kv;
        }
        r.x = out[0];
        r.y = out[1];
    }
    ((float2*)Bm)[k * 2048 + n] = r;
}

// ---------------------------------------------------------------------------
// Inputs: 0 weights[64,256] 1 grid[2048,2] 2 ff_w1 3 ff_b1 4 ff_w2 5 ff_b2
//         6 ff_w3 7 ff_b3 8 k_w1 9 k_b1 10 k_w2 11 k_b2 12 k_w3 13 k_b3
// ---------------------------------------------------------------------------
extern "C" void kernel_launch(void* const* d_in, const int* in_sizes, int n_in,
                              void* d_out, int out_size, void* d_ws, size_t ws_size,
                              hipStream_t stream) {
    const float* weights = (const float*)d_in[0];
    const float* grid    = (const float*)d_in[1];
    const float* ff_w1   = (const float*)d_in[2];
    const float* ff_b1   = (const float*)d_in[3];
    const float* ff_w2   = (const float*)d_in[4];
    const float* ff_b2   = (const float*)d_in[5];
    const float* ff_w3   = (const float*)d_in[6];
    const float* ff_b3   = (const float*)d_in[7];
    const float* k_w1    = (const float*)d_in[8];
    const float* k_b1    = (const float*)d_in[9];
    const float* k_w2    = (const float*)d_in[10];
    const float* k_b2    = (const float*)d_in[11];
    const float* k_w3    = (const float*)d_in[12];
    const float* k_b3    = (const float*)d_in[13];

    float* ws    = (float*)d_ws;
    float* h1    = ws;             // [64 x 128]  (cols 120..127 computed as 0)
    float* h2    = ws + 8192;      // [64 x 240]
    float* integ = ws + 23552;     // [64 x 400]
    float* Bm    = ws + 49152;     // [400 x 4096]
    float* out   = (float*)d_out;  // [64 x 2048 x 2] fp32

    // kval: 819200 pairs, 1 thread each
    kval_kernel<<<3200, 256, 0, stream>>>(grid, k_w1, k_b1, k_w2, k_b2, k_w3, k_b3, Bm);

    // FF layer 1: [64,256]x[256,120] -> h1[64,128], tanh (32 waves, masked N)
    gemm_wmma_f32_masked<<<4, 256, 0, stream>>>(weights, ff_w1, ff_b1, h1,
                                                64, 128, 256, 120, 256, 120, /*act=*/1);
    // FF layer 2: h1[64,128(pad0)]x[120,240] -> h2[64,240], tanh (60 waves, masked K)
    gemm_wmma_f32_masked<<<8, 256, 0, stream>>>(h1, ff_w2, ff_b2, h2,
                                                64, 240, 128, 240, 120, 240, /*act=*/1);
    // FF layer 3: h2[64,240]x[240,400] -> integ[64,400]  (25 waves, unmasked m64)
    gemm_wmma_f32_m64<<<4, 256, 0, stream>>>(h2, ff_w3, ff_b3, integ,
                                             /*N=*/400, /*K=*/240, /*act=*/0);
    // Spread: integ[64,400] x Bm[400,4096] -> sigmoid -> out (256 waves, unmasked m64)
    gemm_wmma_f32_m64<<<32, 256, 0, stream>>>(integ, Bm, nullptr, out,
                                              /*N=*/4096, /*K=*/400, /*act=*/2);
}